// ATripletMarginLossOHNMDM_84808424226946
// MI455X (gfx1250) — compile-verified
//
#include <hip/hip_runtime.h>
#include <math.h>

#define KTOP 3
#define TAU 0.1f
#define MARGIN_MIN 0.1f
#define MARGIN_MAX 0.3f
#define NEG_FILL -50.0f

#define TPB 256
#define CHUNK 1024           // columns per staged chunk = TPB * 4 floats
#define WAVES (TPB / 32)

// Branchless insert of v into descending top-3 (m0 >= m1 >= m2):
// 5 min/max ops -> v_max_f32 / v_min_f32, no control flow.
__device__ __forceinline__ void ins_top3(float v, float& m0, float& m1, float& m2) {
    float t1 = fminf(m0, v);
    m0 = fmaxf(m0, v);
    float t3 = fminf(m1, t1);
    m1 = fmaxf(m1, t1);
    m2 = fmaxf(m2, t3);
}

__global__ __launch_bounds__(TPB) void row_topk_loss_kernel(
    const float* __restrict__ inp, const int* __restrict__ tgt,
    float* __restrict__ row_loss, int B)
{
    __shared__ __align__(16) float s_in[2][CHUNK];
    __shared__ __align__(16) int   s_tg[2][CHUNK];
    __shared__ float s_red[WAVES][3];

    const int row = blockIdx.x;
    const int t   = threadIdx.x;
    const float* rowp = inp + (size_t)row * (size_t)B;
    const int*   tgp  = tgt + (size_t)row * (size_t)B;

    const int nchunks = B / CHUNK;                  // 8 for B=8192
    const unsigned lane_goff = (unsigned)(t * 16);  // 16B per lane inside a chunk

    // ---- stage chunk 0 via async global->LDS copies (gfx1250 ASYNCcnt path) ----
    {
        unsigned la_in = (unsigned)(size_t)(void*)&s_in[0][4 * t];
        unsigned la_tg = (unsigned)(size_t)(void*)&s_tg[0][4 * t];
        unsigned go = lane_goff;
        asm volatile("global_load_async_to_lds_b128 %0, %1, %2"
                     :: "v"(la_in), "v"(go), "s"(rowp) : "memory");
        asm volatile("global_load_async_to_lds_b128 %0, %1, %2"
                     :: "v"(la_tg), "v"(go), "s"(tgp) : "memory");
    }

    float m0 = NEG_FILL, m1 = NEG_FILL, m2 = NEG_FILL;

    for (int j = 0; j < nchunks; ++j) {
        const int buf = j & 1;
        if (j + 1 < nchunks) {
            // stage next chunk into the other buffer, then wait for current one
            const int nbuf = buf ^ 1;
            unsigned la_in = (unsigned)(size_t)(void*)&s_in[nbuf][4 * t];
            unsigned la_tg = (unsigned)(size_t)(void*)&s_tg[nbuf][4 * t];
            unsigned go = (unsigned)((j + 1) * CHUNK * 4) + lane_goff;
            asm volatile("global_load_async_to_lds_b128 %0, %1, %2"
                         :: "v"(la_in), "v"(go), "s"(rowp) : "memory");
            asm volatile("global_load_async_to_lds_b128 %0, %1, %2"
                         :: "v"(la_tg), "v"(go), "s"(tgp) : "memory");
            asm volatile("s_wait_asynccnt 2" ::: "memory");  // chunk j landed
        } else {
            asm volatile("s_wait_asynccnt 0" ::: "memory");  // last chunk landed
        }

        // Each thread consumes exactly the 16B it staged: no cross-wave barrier needed.
        const float4 v = *reinterpret_cast<const float4*>(&s_in[buf][4 * t]);
        const int4   g = *reinterpret_cast<const int4*>(&s_tg[buf][4 * t]);
        // Branchless mask: positives become NEG_FILL, which never beats a real top-3.
        float vx = (g.x == 0) ? v.x : NEG_FILL;
        float vy = (g.y == 0) ? v.y : NEG_FILL;
        float vz = (g.z == 0) ? v.z : NEG_FILL;
        float vw = (g.w == 0) ? v.w : NEG_FILL;
        ins_top3(vx, m0, m1, m2);
        ins_top3(vy, m0, m1, m2);
        ins_top3(vz, m0, m1, m2);
        ins_top3(vw, m0, m1, m2);
    }

    // ---- wave32 reduction: merge top-3 triples across lanes ----
    for (int off = 16; off >= 1; off >>= 1) {
        float a0 = __shfl_xor(m0, off, 32);
        float a1 = __shfl_xor(m1, off, 32);
        float a2 = __shfl_xor(m2, off, 32);
        ins_top3(a0, m0, m1, m2);
        ins_top3(a1, m0, m1, m2);
        ins_top3(a2, m0, m1, m2);
    }

    const int wave = t >> 5;
    if ((t & 31) == 0) {
        s_red[wave][0] = m0; s_red[wave][1] = m1; s_red[wave][2] = m2;
    }
    __syncthreads();

    if (t == 0) {
        float b0 = s_red[0][0], b1 = s_red[0][1], b2 = s_red[0][2];
        #pragma unroll
        for (int w = 1; w < WAVES; ++w) {
            ins_top3(s_red[w][0], b0, b1, b2);
            ins_top3(s_red[w][1], b0, b1, b2);
            ins_top3(s_red[w][2], b0, b1, b2);
        }

        const float sim_p = rowp[row];  // diagonal positive
        float sn[KTOP] = { b0, b1, b2 };
        float l[KTOP], sm[KTOP];
        #pragma unroll
        for (int k = 0; k < KTOP; ++k) {
            float d = fabsf(sim_p - sn[k]);
            d = fminf(fmaxf(d, MARGIN_MIN), MARGIN_MAX);
            l[k]  = fmaxf(0.0f, sn[k] - sim_p + d);
            sm[k] = (l[k] == 0.0f) ? NEG_FILL : sn[k];
        }
        // softmax over K with temperature TAU (numerically stabilized)
        float mx = fmaxf(sm[0], fmaxf(sm[1], sm[2]));
        float e0 = __expf((sm[0] - mx) * (1.0f / TAU));
        float e1 = __expf((sm[1] - mx) * (1.0f / TAU));
        float e2 = __expf((sm[2] - mx) * (1.0f / TAU));
        float inv = 1.0f / (e0 + e1 + e2);
        row_loss[row] = (l[0] * e0 + l[1] * e1 + l[2] * e2) * inv;
    }
}

// Deterministic fixed-order final reduction (no float atomics).
__global__ __launch_bounds__(256) void final_reduce_kernel(
    const float* __restrict__ part, float* __restrict__ out, int n, float scale)
{
    __shared__ float s[256];
    float acc = 0.0f;
    for (int i = threadIdx.x; i < n; i += 256) acc += part[i];
    s[threadIdx.x] = acc;
    __syncthreads();
    for (int off = 128; off > 0; off >>= 1) {
        if ((int)threadIdx.x < off) s[threadIdx.x] += s[threadIdx.x + off];
        __syncthreads();
    }
    if (threadIdx.x == 0) out[0] = s[0] * scale;
}

extern "C" void kernel_launch(void* const* d_in, const int* in_sizes, int n_in,
                              void* d_out, int out_size, void* d_ws, size_t ws_size,
                              hipStream_t stream) {
    const float* inp = (const float*)d_in[0];
    const int*   tgt = (const int*)d_in[1];
    float* out = (float*)d_out;

    const long long n = (long long)in_sizes[0];
    const int B = (int)(sqrt((double)n) + 0.5);      // 8192

    float* rowloss = (float*)d_ws;                   // B floats of scratch

    row_topk_loss_kernel<<<B, TPB, 0, stream>>>(inp, tgt, rowloss, B);
    final_reduce_kernel<<<1, 256, 0, stream>>>(rowloss, out, B,
                                               1.0f / (float)(B * KTOP));
}